// Attention_82583631167867
// MI455X (gfx1250) — compile-verified
//
#include <hip/hip_runtime.h>
#include <hip/hip_bf16.h>
#include <math.h>

// Problem constants (match reference)
#define N_NODES   100000
#define N_EDGES   1600000
#define DIM       128
#define EIG_DIM   32
#define EPS_F     1e-8f
#define INV_SQRT_D 0.08838834764831845f  // 1/sqrt(128)

typedef __attribute__((ext_vector_type(2))) float v2f;
typedef __attribute__((ext_vector_type(8))) float v8f;

// ---------------- helpers ----------------

__device__ __forceinline__ void atomic_add_f32(float* p, float v) {
  // relaxed agent-scope fp32 add -> global_atomic_add_f32
  __hip_atomic_fetch_add(p, v, __ATOMIC_RELAXED, __HIP_MEMORY_SCOPE_AGENT);
}

__device__ __forceinline__ void atomic_max_f32(float* addr, float val) {
  // classic signed/unsigned bit trick; addr initialized to -inf
  if (val >= 0.0f)
    atomicMax((int*)addr, __float_as_int(val));
  else
    atomicMin((unsigned int*)addr, __float_as_uint(val));
}

__device__ __forceinline__ float v8_get(v8f v, int i) {
  switch (i) {
    case 0: return v[0]; case 1: return v[1];
    case 2: return v[2]; case 3: return v[3];
    case 4: return v[4]; case 5: return v[5];
    case 6: return v[6]; default: return v[7];
  }
}

__device__ __forceinline__ float compute_gsim(const float* sums, const float* gamma) {
  float sab = sums[0], saa = sums[1], sbb = sums[2];
  float sim = sab / (fmaxf(sqrtf(saa), EPS_F) * fmaxf(sqrtf(sbb), EPS_F));
  return gamma[0] * sim;
}

// ---------------- kernel 0: init ----------------
// zero d_out (N*DIM), set m0/m1 to -inf, z0/z1 to 0, sums to 0
__global__ void k_init(float* __restrict__ out,
                       float* __restrict__ m0, float* __restrict__ m1,
                       float* __restrict__ z0, float* __restrict__ z1,
                       float* __restrict__ sums) {
  int i = blockIdx.x * blockDim.x + threadIdx.x;
  if (i < N_NODES * DIM) out[i] = 0.0f;
  if (i < N_NODES) {
    m0[i] = -INFINITY;
    m1[i] = -INFINITY;
    z0[i] = 0.0f;
    z1[i] = 0.0f;
  }
  if (i < 3) sums[i] = 0.0f;
}

// ---------------- kernel 1: global motif-cosine reduction ----------------
// sums[0]=sum(a*b), sums[1]=sum(a*a), sums[2]=sum(b*b) over all edges,
// where a = me[row[e]], b = me[col[e]], me[i] = motif_w[motif_ids[i]]
__global__ void k_sim(const int* __restrict__ row, const int* __restrict__ col,
                      const int* __restrict__ motif_ids,
                      const float* __restrict__ motif_w,
                      float* __restrict__ sums) {
  float ab = 0.0f, aa = 0.0f, bb = 0.0f;
  for (int e = blockIdx.x * blockDim.x + threadIdx.x; e < N_EDGES;
       e += gridDim.x * blockDim.x) {
    float a = motif_w[motif_ids[row[e]]];
    float b = motif_w[motif_ids[col[e]]];
    ab = fmaf(a, b, ab);
    aa = fmaf(a, a, aa);
    bb = fmaf(b, b, bb);
  }
  // wave32 butterfly reduce
  #pragma unroll
  for (int m = 16; m > 0; m >>= 1) {
    ab += __shfl_xor(ab, m, 32);
    aa += __shfl_xor(aa, m, 32);
    bb += __shfl_xor(bb, m, 32);
  }
  if ((threadIdx.x & 31) == 0) {
    atomic_add_f32(&sums[0], ab);
    atomic_add_f32(&sums[1], aa);
    atomic_add_f32(&sums[2], bb);
  }
}

// ---------------- kernel 2: per-edge scores via fp32 WMMA ----------------
// Each wave handles a tile of 16 edges. A = q[row[e]] rows (16xK), B = k[col[e]]
// rows acting as columns (Kx16). Diagonal of A*B^T gives the 16 edge dots,
// computed exactly in fp32 with V_WMMA_F32_16X16X4_F32 chained over K.
__global__ void k_edge_scores(const float* __restrict__ q,
                              const float* __restrict__ kk,
                              const float* __restrict__ eigs,
                              const float* __restrict__ mav,
                              const float* __restrict__ lambda0,
                              const float* __restrict__ gamma,
                              const int* __restrict__ row,
                              const int* __restrict__ col,
                              float* __restrict__ s0_out,
                              float* __restrict__ m0,
                              float* __restrict__ m1,
                              const float* __restrict__ sums) {
  const int lane = threadIdx.x & 31;
  const int wavesPerBlock = blockDim.x >> 5;
  const int wave = blockIdx.x * wavesPerBlock + (threadIdx.x >> 5);
  const int nWaves = gridDim.x * wavesPerBlock;
  const int nTiles = N_EDGES / 16;

  const float gs = compute_gsim(sums, gamma);
  const float el0 = expf(lambda0[0]);

  const int m = lane & 15;            // edge-in-tile handled by this lane
  const int koff = (lane >> 4) << 1;  // lanes 0-15 -> K{0,1}; lanes 16-31 -> K{2,3}

  // diagonal ownership: lanes 0..7 hold diag j=lane in acc[lane];
  // lanes 24..31 hold diag j=lane-16 in acc[lane-24]
  const bool hold = (lane < 8) || (lane >= 24);
  const int vi = (lane < 8) ? lane : (lane - 24);

  for (int tile = wave; tile < nTiles; tile += nWaves) {
    const int e = tile * 16 + m;
    const int r = row[e];
    const int c = col[e];
    const float* qp = q    + (size_t)r * DIM + koff;
    const float* kp = kk   + (size_t)c * DIM + koff;
    const float* ep = eigs + (size_t)r * EIG_DIM + koff;
    const float* fp = eigs + (size_t)c * EIG_DIM + koff;

    v8f acc = {};
    #pragma unroll 8
    for (int k0 = 0; k0 < DIM; k0 += 4) {
      v2f a = *(const v2f*)(qp + k0);   // global_load_b64
      v2f b = *(const v2f*)(kp + k0);
      acc = __builtin_amdgcn_wmma_f32_16x16x4_f32(
          false, a, false, b, (short)0, acc, false, false);
    }

    v8f acc2 = {};
    #pragma unroll
    for (int k0 = 0; k0 < EIG_DIM; k0 += 4) {
      v2f a = *(const v2f*)(ep + k0);
      v2f b = *(const v2f*)(fp + k0);
      acc2 = __builtin_amdgcn_wmma_f32_16x16x4_f32(
          false, a, false, b, (short)0, acc2, false, false);
    }

    const float dqk = v8_get(acc, vi);
    const float dee = v8_get(acc2, vi);
    if (hold) {
      // lane's own r is row[e_j] because (lane&15)==j for holder lanes
      const float s0v = fmaf(dqk, INV_SQRT_D, el0 * dee);
      s0_out[e] = s0v;
      atomic_max_f32(&m0[r], s0v);
      const float s1v = mav[e] * gs;
      atomic_max_f32(&m1[r], s1v);
    }
  }
}

// ---------------- kernel 3: segment sums of exponentials ----------------
__global__ void k_znorm(const float* __restrict__ s0,
                        const float* __restrict__ mav,
                        const int* __restrict__ row,
                        const float* __restrict__ m0,
                        const float* __restrict__ m1,
                        float* __restrict__ z0, float* __restrict__ z1,
                        const float* __restrict__ gamma,
                        const float* __restrict__ sums) {
  const float gs = compute_gsim(sums, gamma);
  for (int e = blockIdx.x * blockDim.x + threadIdx.x; e < N_EDGES;
       e += gridDim.x * blockDim.x) {
    const int r = row[e];
    atomic_add_f32(&z0[r], expf(s0[e] - m0[r]));
    atomic_add_f32(&z1[r], expf(mav[e] * gs - m1[r]));
  }
}

// ---------------- kernel 4: normalize + scatter SpMM ----------------
// One wave per edge: 128-wide v[col] read (b128/lane), 4 fp32 atomics/lane to out[row].
__global__ void k_scatter(const float* __restrict__ v,
                          const float* __restrict__ s0,
                          const float* __restrict__ mav,
                          const int* __restrict__ row,
                          const int* __restrict__ col,
                          const float* __restrict__ m0,
                          const float* __restrict__ m1,
                          const float* __restrict__ z0,
                          const float* __restrict__ z1,
                          const float* __restrict__ gamma,
                          const float* __restrict__ sums,
                          float* __restrict__ out) {
  const int lane = threadIdx.x & 31;
  const int wave = (blockIdx.x * blockDim.x + threadIdx.x) >> 5;
  const int nWaves = (gridDim.x * blockDim.x) >> 5;
  const float gs = compute_gsim(sums, gamma);

  for (int e = wave; e < N_EDGES; e += nWaves) {
    const int r = row[e];
    const int c = col[e];
    const float a0 = expf(s0[e] - m0[r]) / z0[r];
    const float a1 = expf(mav[e] * gs - m1[r]) / z1[r];
    const float attn = 0.5f * (a0 + a1);

    const float4 vv = *(const float4*)(v + (size_t)c * DIM + lane * 4);
    float* op = out + (size_t)r * DIM + lane * 4;
    atomic_add_f32(op + 0, attn * vv.x);
    atomic_add_f32(op + 1, attn * vv.y);
    atomic_add_f32(op + 2, attn * vv.z);
    atomic_add_f32(op + 3, attn * vv.w);
  }
}

// ---------------- launch ----------------
extern "C" void kernel_launch(void* const* d_in, const int* in_sizes, int n_in,
                              void* d_out, int out_size, void* d_ws, size_t ws_size,
                              hipStream_t stream) {
  const float* q        = (const float*)d_in[0];
  const float* k        = (const float*)d_in[1];
  const float* v        = (const float*)d_in[2];
  const float* eigs     = (const float*)d_in[3];
  const float* mav      = (const float*)d_in[4];   // motif_adj_vals [E]
  const float* lambda0  = (const float*)d_in[5];
  const float* gamma    = (const float*)d_in[6];
  const float* motif_w  = (const float*)d_in[7];   // [64,1]
  const int*   row      = (const int*)d_in[8];
  const int*   col      = (const int*)d_in[9];
  const int*   motif_id = (const int*)d_in[10];
  float* out = (float*)d_out;

  // workspace layout (floats)
  float* ws   = (float*)d_ws;
  float* s0   = ws;                           // E
  float* m0   = ws + N_EDGES;                 // N
  float* m1   = m0 + N_NODES;                 // N
  float* z0   = m1 + N_NODES;                 // N
  float* z1   = z0 + N_NODES;                 // N
  float* sums = z1 + N_NODES;                 // 3

  // 0) init
  {
    int total = N_NODES * DIM;
    int blocks = (total + 255) / 256;
    k_init<<<blocks, 256, 0, stream>>>(out, m0, m1, z0, z1, sums);
  }
  // 1) global cosine-sim reduction
  k_sim<<<2048, 256, 0, stream>>>(row, col, motif_id, motif_w, sums);
  // 2) per-edge scores via WMMA (8 waves/block, 1 tile of 16 edges per wave-iter)
  {
    int nTiles = N_EDGES / 16;                 // 100000
    int blocks = (nTiles + 7) / 8;             // exactly 1 tile per wave
    k_edge_scores<<<blocks, 256, 0, stream>>>(q, k, eigs, mav, lambda0, gamma,
                                              row, col, s0, m0, m1, sums);
  }
  // 3) segment sums
  k_znorm<<<(N_EDGES + 255) / 256, 256, 0, stream>>>(s0, mav, row, m0, m1,
                                                     z0, z1, gamma, sums);
  // 4) scatter SpMM
  k_scatter<<<8192, 256, 0, stream>>>(v, s0, mav, row, col, m0, m1,
                                      z0, z1, gamma, sums, out);
}